// IndexNet_42786464202885
// MI455X (gfx1250) — compile-verified
//
#include <hip/hip_runtime.h>
#include <stdint.h>

typedef float v2f __attribute__((ext_vector_type(2)));
typedef float v8f __attribute__((ext_vector_type(8)));

#define NROWS 131072
#define DDIM  3
#define HDIM  128
#define ROWS_PER_WAVE  16
#define WAVES_PER_BLK  4
#define ROWS_PER_BLK   (ROWS_PER_WAVE * WAVES_PER_BLK)   // 64
#define NTHREADS       (WAVES_PER_BLK * 32)              // 128

__global__ __launch_bounds__(NTHREADS)
void indexnet_mlp_wmma(const float* __restrict__ x,
                       const uint8_t* __restrict__ mask,
                       const float* __restrict__ W0, const float* __restrict__ b0,
                       const float* __restrict__ W1, const float* __restrict__ b1,
                       const float* __restrict__ W2, const float* __restrict__ b2,
                       const float* __restrict__ W3, const float* __restrict__ b3,
                       float* __restrict__ out)
{
    // Per-wave activation slabs (16 rows x 128 cols, fp32): 4 * 8KB = 32 KB static
    __shared__ float sAct[WAVES_PER_BLK * ROWS_PER_WAVE * HDIM];
    // One layer's weights, K-pair interleaved: sW[(k/2)*256 + col*2 + (k&1)]
    // = W[k][col].  64 KB dynamic LDS.
    extern __shared__ float sW[];

    const int lane = threadIdx.x & 31;
    const int wave = threadIdx.x >> 5;
    const int rowBase = blockIdx.x * ROWS_PER_BLK + wave * ROWS_PER_WAVE;
    float* hAct = sAct + wave * ROWS_PER_WAVE * HDIM;

    const int mrow  = lane & 15;        // A: M row; B/C/D: column-within-tile
    const int khalf = (lane >> 4) * 2;  // A/B: K offset for upper half-wave
    const int roff  = (lane >> 4) * 8;  // C/D: M offset for upper half-wave

    v8f oacc[8];
    #pragma unroll
    for (int t = 0; t < 8; ++t)
        oacc[t] = (v8f){0.f,0.f,0.f,0.f,0.f,0.f,0.f,0.f};

    for (int d = 0; d < DDIM; ++d) {
        // ---- layer 0: h = relu(x * W0[d,:] + b0[d,:]) into per-wave slab ----
        #pragma unroll 4
        for (int m = 0; m < ROWS_PER_WAVE; ++m) {
            const float xv = x[(rowBase + m) * DDIM + d];
            #pragma unroll
            for (int j = lane; j < HDIM; j += 32) {
                hAct[m * HDIM + j] =
                    fmaxf(fmaf(xv, W0[d * HDIM + j], b0[d * HDIM + j]), 0.f);
            }
        }

        const float* Wl[3] = { W1 + d * HDIM * HDIM,
                               W2 + d * HDIM * HDIM,
                               W3 + d * HDIM * HDIM };
        const float* bl[3] = { b1 + d * HDIM, b2 + d * HDIM, b3 + d * HDIM };

        for (int layer = 0; layer < 3; ++layer) {
            // Stage weights into LDS, K-pair interleaved so that each B
            // fragment (W[k][col], W[k+1][col]) is one contiguous 8B element.
            __syncthreads();   // previous users of sW are done
            {
                const float* src = Wl[layer];
                // q = p*128 + col, p = K-pair index (0..63)
                for (int q = threadIdx.x; q < (HDIM / 2) * HDIM; q += NTHREADS) {
                    const int col = q & (HDIM - 1);
                    const int p   = q >> 7;
                    v2f w;
                    w.x = src[(2 * p)     * HDIM + col];
                    w.y = src[(2 * p + 1) * HDIM + col];
                    *(v2f*)(sW + p * (2 * HDIM) + col * 2) = w;
                }
            }
            __syncthreads();   // weights visible to all waves

            // ---- GEMM: [16 x 128] = hAct[16 x 128] * W[128 x 128] ----
            v8f c[8];
            #pragma unroll
            for (int t = 0; t < 8; ++t)
                c[t] = (v8f){0.f,0.f,0.f,0.f,0.f,0.f,0.f,0.f};

            #pragma unroll 4
            for (int kk = 0; kk < HDIM / 4; ++kk) {
                const int k0 = kk * 4;
                // A frag: lanes 0-15 -> K=k0,k0+1; lanes 16-31 -> K=k0+2,k0+3
                const v2f a = *(const v2f*)(hAct + mrow * HDIM + k0 + khalf);
                // interleaved base for this K pair: (k0+khalf)/2 * 256
                const float* wbase = sW + (k0 + khalf) * HDIM;
                #pragma unroll
                for (int t = 0; t < 8; ++t) {
                    const int col = t * 16 + mrow;
                    const v2f bfrag = *(const v2f*)(wbase + col * 2);
                    c[t] = __builtin_amdgcn_wmma_f32_16x16x4_f32(
                        /*neg_a=*/false, a, /*neg_b=*/false, bfrag,
                        /*c_mod=*/(short)0, c[t],
                        /*reuse_a=*/false, /*reuse_b=*/false);
                }
            }

            const float* bias = bl[layer];
            if (layer < 2) {
                // bias + ReLU, transpose D-layout back into A-layout via LDS slab
                #pragma unroll
                for (int t = 0; t < 8; ++t) {
                    const float bb = bias[t * 16 + mrow];
                    #pragma unroll
                    for (int r = 0; r < 8; ++r) {
                        const float v = fmaxf(c[t][r] + bb, 0.f);
                        hAct[(r + roff) * HDIM + t * 16 + mrow] = v;
                    }
                }
            } else {
                // final layer: bias, mask, accumulate into output registers
                float mk[8];
                #pragma unroll
                for (int r = 0; r < 8; ++r)
                    mk[r] = mask[(rowBase + r + roff) * DDIM + d] ? 1.f : 0.f;
                #pragma unroll
                for (int t = 0; t < 8; ++t) {
                    const float bb = bias[t * 16 + mrow];
                    #pragma unroll
                    for (int r = 0; r < 8; ++r)
                        oacc[t][r] += mk[r] * (c[t][r] + bb);
                }
            }
        }
    }

    // ---- store [16 x 128] per wave; each half-wave writes 64B segments ----
    #pragma unroll
    for (int t = 0; t < 8; ++t) {
        #pragma unroll
        for (int r = 0; r < 8; ++r)
            out[(rowBase + r + roff) * HDIM + t * 16 + mrow] = oacc[t][r];
    }
}

extern "C" void kernel_launch(void* const* d_in, const int* in_sizes, int n_in,
                              void* d_out, int out_size, void* d_ws, size_t ws_size,
                              hipStream_t stream)
{
    (void)in_sizes; (void)n_in; (void)d_ws; (void)ws_size; (void)out_size;

    const float*   x    = (const float*)  d_in[0];
    const uint8_t* mask = (const uint8_t*)d_in[1];   // jax bool -> 1 byte/elem
    const float*   W0   = (const float*)  d_in[2];
    const float*   b0   = (const float*)  d_in[3];
    const float*   W1   = (const float*)  d_in[4];
    const float*   b1   = (const float*)  d_in[5];
    const float*   W2   = (const float*)  d_in[6];
    const float*   b2   = (const float*)  d_in[7];
    const float*   W3   = (const float*)  d_in[8];
    const float*   b3   = (const float*)  d_in[9];
    float* out = (float*)d_out;

    const int grid = NROWS / ROWS_PER_BLK;              // 2048 blocks
    const size_t dyn_lds = HDIM * HDIM * sizeof(float); // 64 KB weight stage
    indexnet_mlp_wmma<<<grid, NTHREADS, dyn_lds, stream>>>(
        x, mask, W0, b0, W1, b1, W2, b2, W3, b3, out);
}